// MessagePassing_85572928405766
// MI455X (gfx1250) — compile-verified
//
#include <hip/hip_runtime.h>
#include <stdint.h>

// Scatter-add message passing: out[dst[e], :] += x[src[e], :], d=32, E=1.6M.
//
// MI455X plan:
//  * Bound by L2 f32-atomic throughput (51.2M atomics); x (12.8MB) and out
//    (12.8MB) are L2-resident (192MB L2), HBM traffic ~40MB (~2us @ 23.3TB/s).
//  * 8 lanes per edge, one float4 per lane -> each row is one coalesced 128B
//    global_load_b128 group; 4 no-return global_atomic_add_f32 per lane.
//  * Per-wave tile of 128 edges: (src,dst) index pairs staged interleaved into
//    LDS via CDNA5 async global->LDS copies (ASYNCcnt) + s_wait_asynccnt,
//    then read back with one ds_load_b64 per edge (broadcast across octet).
//  * All index math in 32-bit (everything < 2^31) to avoid i64 VALU chains.

#define WAVE   32
#define WTILE  128                 // edges staged per wave
#define BLOCK  256                 // 8 waves per block
#define WPB    (BLOCK / WAVE)
#define BTILE  (WTILE * WPB)       // 1024 edges per block

__global__ void __launch_bounds__(BLOCK)
scatter_add_kernel(const float* __restrict__ x,
                   const int* __restrict__ ei,   // [2, E] int32 (src row, dst row)
                   float* __restrict__ out,
                   int E)
{
    __shared__ int s_idx[2 * BTILE];   // per wave: 128 (src,dst) pairs = 1KB

    const int lane = threadIdx.x & (WAVE - 1);
    const int wv   = threadIdx.x >> 5;

    int* s_pairs = &s_idx[wv * (2 * WTILE)];             // {src,dst} interleaved
    const unsigned lds_base = (unsigned)(uintptr_t)s_pairs;

    const int e0 = (int)blockIdx.x * BTILE + wv * WTILE; // first edge of wave tile

    // ---- Stage 128 (src,dst) index pairs into LDS via async global->LDS. ----
#pragma unroll
    for (int k = 0; k < WTILE / WAVE; ++k) {
        const int el = k * WAVE + lane;                  // local edge 0..127
        int e = e0 + el;
        if (e >= E) e = 0;                               // clamp to safe address
        const int* gsrc = ei + e;
        const int* gdst = ei + E + e;
        const unsigned lsrc = lds_base + (unsigned)(el * 8);
        const unsigned ldst = lsrc + 4;
        asm volatile("global_load_async_to_lds_b32 %0, %1, off"
                     :: "v"(lsrc), "v"(gsrc) : "memory");
        asm volatile("global_load_async_to_lds_b32 %0, %1, off"
                     :: "v"(ldst), "v"(gdst) : "memory");
    }
    asm volatile("s_wait_asynccnt 0" ::: "memory");

    // ---- Process: 8 lanes per edge; chunk c = lane&7 is loop-invariant. ----
    const int c4 = (lane & 7) * 4;            // feature offset of this lane
    const float* xb = x + c4;                 // lane-fixed column base
    float*       ob = out + c4;
    const int2* sp = (const int2*)s_pairs;

    if (e0 + WTILE <= E) {
        // Full tile: no bounds checks.
#pragma unroll 8
        for (int el = (lane >> 3); el < WTILE; el += 4) {
            const int2 sd = sp[el];                      // one ds_load_b64
            const float4 v = *(const float4*)(xb + sd.x * 32);
            float* o = ob + sd.y * 32;
            unsafeAtomicAdd(o + 0, v.x);   // global_atomic_add_f32, no return
            unsafeAtomicAdd(o + 1, v.y);
            unsafeAtomicAdd(o + 2, v.z);
            unsafeAtomicAdd(o + 3, v.w);
        }
    } else {
        // Tail tile: per-edge check (32-bit compare).
#pragma unroll 4
        for (int el = (lane >> 3); el < WTILE; el += 4) {
            if (e0 + el < E) {
                const int2 sd = sp[el];
                const float4 v = *(const float4*)(xb + sd.x * 32);
                float* o = ob + sd.y * 32;
                unsafeAtomicAdd(o + 0, v.x);
                unsafeAtomicAdd(o + 1, v.y);
                unsafeAtomicAdd(o + 2, v.z);
                unsafeAtomicAdd(o + 3, v.w);
            }
        }
    }
}

extern "C" void kernel_launch(void* const* d_in, const int* in_sizes, int n_in,
                              void* d_out, int out_size, void* d_ws, size_t ws_size,
                              hipStream_t stream)
{
    const float* x  = (const float*)d_in[0];   // [100000, 32] f32
    const int*   ei = (const int*)d_in[1];     // [2, E] int32
    float*       out = (float*)d_out;          // [100000, 32] f32
    const int E = in_sizes[1] / 2;

    // Deterministic accumulation base (d_out is poisoned, not re-zeroed).
    hipMemsetAsync(d_out, 0, (size_t)out_size * sizeof(float), stream);

    const int blocks = (E + BTILE - 1) / BTILE;
    scatter_add_kernel<<<blocks, BLOCK, 0, stream>>>(x, ei, out, E);
}